// Propagation_66735201845639
// MI455X (gfx1250) — compile-verified
//
#include <hip/hip_runtime.h>

// Propagation: out[b, d*3+o, h, w] = x[b, d, h, w + o - 1], zero-padded along W.
// Zero-FLOP streaming op (~503 MB traffic -> ~21.6us @ 23.3 TB/s). Strategy:
//  - each WAVE owns whole rows: halo reads stay wave-private, NO barriers
//  - double-buffered async DMA pipeline: row i+1 global->LDS in flight
//    (ASYNCcnt, in-order done => s_wait_asynccnt 8) while row i is consumed
//  - LDS realigns the +/-1 shifts; 16B-aligned non-temporal b128 stores
//    (output 360MiB > L2, never re-read)

typedef float v4f __attribute__((ext_vector_type(4)));

constexpr int W    = 960;              // row length
constexpr int H    = 512;
constexpr int BD   = 2 * 16;           // B * D
constexpr int ROWS = BD * H;           // rows per input = 16384
constexpr int W4   = W / 4;            // 240 b128 chunks per row
constexpr int F    = 3;
constexpr int RPW  = 4;                // rows per wave (divides H and ROWS)
constexpr int WPB  = 8;                // waves per block (256 threads)
constexpr long long N = (long long)ROWS * W;   // elements per input

__global__ __launch_bounds__(256) void prop_kernel(const float* __restrict__ x,
                                                   const float* __restrict__ y,
                                                   float* __restrict__ out) {
    __shared__ __align__(16) float lds[WPB][2][W];   // 2 row buffers per wave

    const int lane = threadIdx.x & 31;
    const int wv   = threadIdx.x >> 5;
    const int gw   = blockIdx.x * WPB + wv;          // global wave id
    const int t0   = gw * RPW;                       // first global row
    const int s    = (t0 >= ROWS) ? 1 : 0;           // 0 = offset_x, 1 = offset_y
    const int r0   = t0 - s * ROWS;
    const float* __restrict__ src = s ? y : x;
    float* __restrict__ dst = out + (long long)s * F * N;

    const int bd = r0 / H;            // constant across the RPW rows (RPW | H)
    const int h0 = r0 - bd * H;

    // Issue one row's DMA: 8 async b128 instructions per wave (7 full + 1 partial).
    auto stage = [&](int i, int b) {
        const float* g = src + (long long)(r0 + i) * W;
        #pragma unroll
        for (int j = 0; j < 8; ++j) {
            const int idx = j * 32 + lane;
            if (idx < W4) {
                unsigned lds_off =
                    (unsigned)(unsigned long long)(const void*)&lds[wv][b][idx * 4];
                unsigned long long ga =
                    (unsigned long long)(const void*)(g + idx * 4);
                asm volatile("global_load_async_to_lds_b128 %0, %1, off"
                             :: "v"(lds_off), "v"(ga)
                             : "memory");
            }
        }
    };

    // Compose the 3 shifted rows from LDS and NT-store them.
    auto consume = [&](int i, int b) {
        const float* L = &lds[wv][b][0];
        const int h = h0 + i;
        float* o0 = dst + ((long long)(bd * F) * H + h) * W;   // o = 0 channel
        float* o1 = o0 + (long long)H * W;                     // o = 1
        float* o2 = o1 + (long long)H * W;                     // o = 2
        #pragma unroll
        for (int j = 0; j < 8; ++j) {
            const int idx = j * 32 + lane;
            if (idx < W4) {
                const int w = idx * 4;
                const v4f v   = *(const v4f*)&L[w];            // ds_load_b128
                const float xm1 = (idx > 0)      ? L[w - 1] : 0.0f;
                const float xp4 = (idx < W4 - 1) ? L[w + 4] : 0.0f;
                v4f a0 = { xm1, v.x, v.y, v.z };   // out = x[w-1]
                v4f a2 = { v.y, v.z, v.w, xp4 };   // out = x[w+1]
                __builtin_nontemporal_store(a0, (v4f*)(o0 + w));
                __builtin_nontemporal_store(v,  (v4f*)(o1 + w));
                __builtin_nontemporal_store(a2, (v4f*)(o2 + w));
            }
        }
    };

    // Double-buffered pipeline: next row's DMA overlaps current row's stores.
    stage(0, 0);
    #pragma unroll
    for (int i = 0; i < RPW - 1; ++i) {
        stage(i + 1, (i + 1) & 1);
        // 16 outstanding; async-load "done" is in-order, so <=8 means row i landed.
        asm volatile("s_wait_asynccnt 0x8" ::: "memory");
        consume(i, i & 1);
    }
    asm volatile("s_wait_asynccnt 0x0" ::: "memory");
    consume(RPW - 1, (RPW - 1) & 1);
}

extern "C" void kernel_launch(void* const* d_in, const int* in_sizes, int n_in,
                              void* d_out, int out_size, void* d_ws, size_t ws_size,
                              hipStream_t stream) {
    (void)in_sizes; (void)n_in; (void)out_size; (void)d_ws; (void)ws_size;
    const float* x = (const float*)d_in[0];
    const float* y = (const float*)d_in[1];
    float* out     = (float*)d_out;

    // 2 inputs * 16384 rows / (4 rows per wave) = 8192 waves = 1024 blocks
    const int total_waves = 2 * ROWS / RPW;
    prop_kernel<<<dim3(total_waves / WPB), dim3(256), 0, stream>>>(x, y, out);
}